// GCNConv_62182536511747
// MI455X (gfx1250) — compile-verified
//
#include <hip/hip_runtime.h>

// ---------------------------------------------------------------------------
// GCNConv for MI455X (gfx1250, wave32):
//   Stage 1: XW = X @ W via V_WMMA_F32_16X16X4_F32. Memory-bound (AI ~32
//            FLOP/B) -> fp32 WMMA saturates HBM and matches reference math.
//            Block = 256 thr (8 waves) covers 32 rows x 128 cols:
//              - X tile (32x128) staged once into LDS, coalesced b128,
//                padded pitch 132 (132%64==4 -> conflict-free frag reads)
//              - wave w owns column tile w; processes BOTH row tiles,
//                reusing each B fragment for 2 WMMAs (64 WMMA/wave).
//   Stage 2: out = A @ XW. edge_rows sorted -> one block per dst row,
//            binary-searched segment, 4-way unrolled register-accumulated
//            gather (4 L2 loads in flight/thread), no atomics.
//            XW (51.2 MB) is resident in the 192 MB L2.
// ---------------------------------------------------------------------------

typedef __attribute__((ext_vector_type(2))) float v2f;
typedef __attribute__((ext_vector_type(8))) float v8f;

#define D_FEAT    128   // D_IN == D_OUT == 128
#define ROWS_PB   32    // rows per block (2 row-tiles of 16)
#define LDS_PITCH 132   // floats; multiple of 4 (b128 stores ok), %64==4

// ---------------------------------------------------------------------------
// Stage 1: dense GEMM with fp32 WMMA (16x16x4), LDS-staged A operand.
// ---------------------------------------------------------------------------
__global__ void __launch_bounds__(256)
gcn_gemm_xw_wmma(const float* __restrict__ X,
                 const float* __restrict__ W,
                 float* __restrict__ XW,
                 int n_nodes) {
    __shared__ float xtile[ROWS_PB * LDS_PITCH];   // 16.5 KB

    const int tid  = threadIdx.x;
    const int lane = tid & 31;
    const int wave = tid >> 5;          // 0..7 -> column tile
    const int half = lane >> 4;         // 0: lanes 0-15, 1: lanes 16-31
    const int l    = lane & 15;

    const int row0 = blockIdx.x * ROWS_PB;
    const int col0 = wave * 16;

    // --- Cooperative coalesced stage: X[row0..row0+32) x 128 -> LDS ------
    // 4096 floats = 1024 float4; 4 per thread. Each wave stores one full
    // row per iteration (lane = float4 column) -> conflict-free ds_store.
#pragma unroll
    for (int i = 0; i < 4; ++i) {
        const int f  = i * 256 + tid;   // float4 index in 32x32 grid
        const int r  = f >> 5;          // 32 float4s per row
        const int c4 = f & 31;
        int gr = row0 + r;
        if (gr >= n_nodes) gr = n_nodes - 1;       // clamp (EXEC stays full)
        const float4 v = *(const float4*)(X + (size_t)gr * D_FEAT + c4 * 4);
        float* dst = &xtile[r * LDS_PITCH + c4 * 4];
        dst[0] = v.x; dst[1] = v.y; dst[2] = v.z; dst[3] = v.w;
    }
    __syncthreads();

    // --- K loop: 32 steps, 2 WMMAs per step (two row tiles share B) ------
    // A 16x4 layout: lane(half,l) holds {K=kk, K=kk+1} of row l -> float2.
    // B 4x16 layout: lane(half,l) holds W[kk][col0+l], W[kk+1][col0+l].
    const float* __restrict__ Wc  = W + col0 + l;
    const float* __restrict__ xa0 = &xtile[(l)      * LDS_PITCH];
    const float* __restrict__ xa1 = &xtile[(16 + l) * LDS_PITCH];

    v8f acc0 = {};
    v8f acc1 = {};

#pragma unroll
    for (int k = 0; k < D_FEAT; k += 4) {
        const int kk = k + 2 * half;
        v2f a0 = *(const v2f*)(xa0 + kk);          // ds_load_b64
        v2f a1 = *(const v2f*)(xa1 + kk);          // ds_load_b64
        v2f b;
        b.x = Wc[(size_t)(kk + 0) * D_FEAT];
        b.y = Wc[(size_t)(kk + 1) * D_FEAT];
        acc0 = __builtin_amdgcn_wmma_f32_16x16x4_f32(
            false, a0, false, b, (short)0, acc0, false, false);
        acc1 = __builtin_amdgcn_wmma_f32_16x16x4_f32(
            false, a1, false, b, (short)0, acc1, false, false);
    }

    // --- Store D tiles: lane(half,l) holds rows M=half*8+v at col col0+l --
    float* __restrict__ Oc = XW + col0 + l;
#pragma unroll
    for (int v = 0; v < 8; ++v) {
        const int m0 = row0 + half * 8 + v;
        const int m1 = m0 + 16;
        if (m0 < n_nodes) Oc[(size_t)m0 * D_FEAT] = acc0[v];
        if (m1 < n_nodes) Oc[(size_t)m1 * D_FEAT] = acc1[v];
    }
}

// ---------------------------------------------------------------------------
// Stage 2: segmented SpMM. One block (128 threads) per destination row.
// Binary search is block-uniform; gather loop unrolled 4x for MLP.
// ---------------------------------------------------------------------------
__global__ void __launch_bounds__(128)
gcn_spmm_rows(const float* __restrict__ XW,
              const int*   __restrict__ edge_rows,
              const int*   __restrict__ edge_cols,
              const float* __restrict__ edge_vals,
              float* __restrict__ out,
              int n_edges) {
    const int r = blockIdx.x;
    const int j = threadIdx.x;   // feature index 0..127

    // lower_bound(edge_rows, r)
    int lo = 0, hi = n_edges;
    while (lo < hi) {
        int mid = (lo + hi) >> 1;
        if (edge_rows[mid] < r) lo = mid + 1; else hi = mid;
    }
    const int seg_begin = lo;
    // lower_bound(edge_rows, r+1)
    hi = n_edges;
    while (lo < hi) {
        int mid = (lo + hi) >> 1;
        if (edge_rows[mid] <= r) lo = mid + 1; else hi = mid;
    }
    const int seg_end = lo;

    float acc0 = 0.0f, acc1 = 0.0f;
    int e = seg_begin;
    // 4-way unroll: 4 independent 512B L2 gathers in flight per thread.
    for (; e + 4 <= seg_end; e += 4) {
        const int   c0 = edge_cols[e + 0], c1 = edge_cols[e + 1];
        const int   c2 = edge_cols[e + 2], c3 = edge_cols[e + 3];
        const float w0 = edge_vals[e + 0], w1 = edge_vals[e + 1];
        const float w2 = edge_vals[e + 2], w3 = edge_vals[e + 3];
        const float x0 = XW[(size_t)c0 * D_FEAT + j];
        const float x1 = XW[(size_t)c1 * D_FEAT + j];
        const float x2 = XW[(size_t)c2 * D_FEAT + j];
        const float x3 = XW[(size_t)c3 * D_FEAT + j];
        acc0 = fmaf(w0, x0, acc0);
        acc1 = fmaf(w1, x1, acc1);
        acc0 = fmaf(w2, x2, acc0);
        acc1 = fmaf(w3, x3, acc1);
    }
    for (; e < seg_end; ++e) {
        acc0 = fmaf(edge_vals[e], XW[(size_t)edge_cols[e] * D_FEAT + j], acc0);
    }
    out[(size_t)r * D_FEAT + j] = acc0 + acc1;   // coalesced; empty rows -> 0
}

// ---------------------------------------------------------------------------
// Host-side launch. Inputs (reference order):
//   d_in[0]=X [N*128] f32, d_in[1]=W [128*128] f32,
//   d_in[2]=edge_rows [E] i32 (sorted), d_in[3]=edge_cols [E] i32,
//   d_in[4]=edge_vals [E] f32.  d_out = [N*128] f32.  d_ws holds XW.
// ---------------------------------------------------------------------------
extern "C" void kernel_launch(void* const* d_in, const int* in_sizes, int n_in,
                              void* d_out, int out_size, void* d_ws, size_t ws_size,
                              hipStream_t stream) {
    const float* X  = (const float*)d_in[0];
    const float* W  = (const float*)d_in[1];
    const int*   er = (const int*)d_in[2];
    const int*   ec = (const int*)d_in[3];
    const float* ev = (const float*)d_in[4];
    float* out = (float*)d_out;
    float* XW  = (float*)d_ws;   // n_nodes * 128 floats (51.2 MB)

    const int n_nodes = in_sizes[0] / D_FEAT;
    const int n_edges = in_sizes[2];

    const int mblocks = (n_nodes + ROWS_PB - 1) / ROWS_PB;
    gcn_gemm_xw_wmma<<<mblocks, 256, 0, stream>>>(X, W, XW, n_nodes);
    gcn_spmm_rows<<<n_nodes, D_FEAT, 0, stream>>>(XW, er, ec, ev, out, n_edges);
}